// QGCNConv_56788057588118
// MI455X (gfx1250) — compile-verified
//
#include <hip/hip_runtime.h>
#include <math.h>

typedef float v2f __attribute__((ext_vector_type(2)));
typedef float v8f __attribute__((ext_vector_type(8)));

#define DID 64  // D_IN == D_OUT == 64

__device__ __forceinline__ float wave_sum(float v) {
#pragma unroll
  for (int off = 16; off > 0; off >>= 1) v += __shfl_xor(v, off, 32);
  return v;
}

__device__ __forceinline__ float artanh_clip(float x) {
  x = fminf(fmaxf(x, -1.0f + 1e-7f), 1.0f - 1e-7f);
  return atanhf(x);
}

// ---------------------------------------------------------------------------
// Kernel 1: mx = x @ W^T using V_WMMA_F32_16X16X4_F32 (fp32 WMMA, wave32).
// Block = 128 threads = 4 waves. Block tile: 16 rows x 64 cols; wave w owns
// the 16x16 tile at columns [16w, 16w+16).
// A (16x4 f32) layout: lanes 0-15 -> M=lane, K=0/1 in the 2 VGPRs;
//                      lanes 16-31 -> M=lane-16, K=2/3.
// B (4x16 f32) layout mirrors A with N across lanes.
// C/D (16x16 f32, v8f): VGPR r: lanes 0-15 -> M=r, lanes 16-31 -> M=r+8.
// ---------------------------------------------------------------------------
__global__ void gemm_wmma_f32(const float* __restrict__ x,
                              const float* __restrict__ W,
                              float* __restrict__ mx, int N) {
  const int lane = threadIdx.x & 31;
  const int wave = threadIdx.x >> 5;  // 0..3
  const int l16  = lane & 15;
  const int half = lane >> 4;         // 0 or 1
  const int row0 = blockIdx.x * 16;
  const int n0   = wave * 16;

  int arow = row0 + l16;
  if (arow >= N) arow = N - 1;  // N % 16 == 0 in practice; keep EXEC all-1s
  const float* __restrict__ xrow = x + (size_t)arow * DID;
  const float* __restrict__ wrow = W + (size_t)(n0 + l16) * DID;  // mx[m][n] = sum_k x[m][k]*W[n][k]

  v8f acc = {};
#pragma unroll
  for (int k0 = 0; k0 < DID; k0 += 4) {
    const int k = k0 + 2 * half;
    v2f a, b;
    a.x = xrow[k];
    a.y = xrow[k + 1];
    b.x = wrow[k];
    b.y = wrow[k + 1];
    acc = __builtin_amdgcn_wmma_f32_16x16x4_f32(false, a, false, b, (short)0,
                                                acc, false, false);
  }

#pragma unroll
  for (int r = 0; r < 8; ++r) {
    const int row = row0 + r + 8 * half;
    if (row < N) mx[(size_t)row * DID + n0 + l16] = acc[r];
  }
}

// ---------------------------------------------------------------------------
// Kernel 2: per-row (wave32 per row, 2 cols/lane) Mobius pipeline:
//   res  = mobius_matvec scaling of mx
//   res  = proj(res); h = proj(mobius_add(res, proj(expmap0(b))))
//   xt   = logmap0(h)
// c = 1 everywhere (C_IN == 1).
// ---------------------------------------------------------------------------
__global__ void post_gemm_rowwise(const float* __restrict__ x,
                                  const float* __restrict__ mx,
                                  const float* __restrict__ b,
                                  float* __restrict__ xt, int N) {
  const int lane = threadIdx.x & 31;
  const int row  = blockIdx.x * (blockDim.x >> 5) + (threadIdx.x >> 5);
  if (row >= N) return;
  const float kEps  = 1e-15f;
  const float maxn  = 1.0f - 4e-3f;  // (1-BALL_EPS)/sqrt(c), c=1

  const float* __restrict__ xr = x + (size_t)row * DID;
  const float* __restrict__ mr = mx + (size_t)row * DID;
  float x0 = xr[lane], x1 = xr[lane + 32];
  float m0 = mr[lane], m1 = mr[lane + 32];

  // mobius_matvec scale
  float xn2 = wave_sum(x0 * x0 + x1 * x1);
  float mn2 = wave_sum(m0 * m0 + m1 * m1);
  float xn  = fmaxf(sqrtf(xn2), kEps);
  float mn  = fmaxf(sqrtf(mn2), kEps);
  float sc  = tanhf(mn / xn * artanh_clip(xn)) / mn;
  float r0 = sc * m0, r1 = sc * m1;
  if (mn2 == 0.0f) { r0 = 0.0f; r1 = 0.0f; }  // all(mx==0) guard

  // proj(res)
  float rn = fmaxf(sqrtf(wave_sum(r0 * r0 + r1 * r1)), kEps);
  if (rn > maxn) { float s = maxn / rn; r0 *= s; r1 *= s; }

  // hyp_b = proj(expmap0(b))
  float b0 = b[lane], b1 = b[lane + 32];
  float bn = fmaxf(sqrtf(wave_sum(b0 * b0 + b1 * b1)), kEps);
  float bs = tanhf(bn) / bn;
  float h0 = bs * b0, h1 = bs * b1;
  float hbn = fmaxf(sqrtf(wave_sum(h0 * h0 + h1 * h1)), kEps);
  if (hbn > maxn) { float s = maxn / hbn; h0 *= s; h1 *= s; }

  // mobius_add(res, hyp_b), c = 1
  float x2 = wave_sum(r0 * r0 + r1 * r1);
  float y2 = wave_sum(h0 * h0 + h1 * h1);
  float xy = wave_sum(r0 * h0 + r1 * h1);
  float cx  = 1.0f + 2.0f * xy + y2;
  float cy  = 1.0f - x2;
  float den = fmaxf(1.0f + 2.0f * xy + x2 * y2, kEps);
  float a0 = (cx * r0 + cy * h0) / den;
  float a1 = (cx * r1 + cy * h1) / den;

  // proj(h)
  float an = fmaxf(sqrtf(wave_sum(a0 * a0 + a1 * a1)), kEps);
  if (an > maxn) { float s = maxn / an; a0 *= s; a1 *= s; }

  // logmap0(h)
  float ln = fmaxf(sqrtf(wave_sum(a0 * a0 + a1 * a1)), kEps);
  float ls = artanh_clip(ln) / ln;
  xt[(size_t)row * DID + lane]      = ls * a0;
  xt[(size_t)row * DID + lane + 32] = ls * a1;
}

// ---------------------------------------------------------------------------
// Kernel 3: zero the accumulator (workspace is poisoned by the harness).
// ---------------------------------------------------------------------------
__global__ void zero_f32(float* __restrict__ p, size_t n) {
  size_t i      = (size_t)blockIdx.x * blockDim.x + threadIdx.x;
  size_t stride = (size_t)gridDim.x * blockDim.x;
  for (; i < n; i += stride) p[i] = 0.0f;
}

// ---------------------------------------------------------------------------
// Kernel 4: SpMM scatter: one wave per edge; 32 lanes x 2 floats cover D=64.
// support[row,:] += val * xt[col,:]  via L2-resident fp32 atomics.
// ---------------------------------------------------------------------------
__global__ void spmm_atomic(const int* __restrict__ erow,
                            const int* __restrict__ ecol,
                            const float* __restrict__ eval,
                            const float* __restrict__ xt,
                            float* __restrict__ support, int E) {
  const int lane = threadIdx.x & 31;
  const int e    = blockIdx.x * (blockDim.x >> 5) + (threadIdx.x >> 5);
  if (e >= E) return;
  const int   row = erow[e];
  const int   col = ecol[e];
  const float v   = eval[e];
  const float* __restrict__ src = xt + (size_t)col * DID;
  float* __restrict__ dst = support + (size_t)row * DID;
  atomicAdd(&dst[lane],      v * src[lane]);
  atomicAdd(&dst[lane + 32], v * src[lane + 32]);
}

// ---------------------------------------------------------------------------
// Kernel 5: clamp -> expmap0 -> proj -> logmap0 -> relu -> expmap0 -> proj.
// ---------------------------------------------------------------------------
__global__ void final_rowwise(const float* __restrict__ support,
                              float* __restrict__ out, int N) {
  const int lane = threadIdx.x & 31;
  const int row  = blockIdx.x * (blockDim.x >> 5) + (threadIdx.x >> 5);
  if (row >= N) return;
  const float kEps = 1e-15f;
  const float maxn = 1.0f - 4e-3f;

  const float* __restrict__ sr = support + (size_t)row * DID;
  float s0 = fminf(sr[lane], 1e6f);
  float s1 = fminf(sr[lane + 32], 1e6f);

  // h = proj(expmap0(support))
  float n = fmaxf(sqrtf(wave_sum(s0 * s0 + s1 * s1)), kEps);
  float es = tanhf(n) / n;
  float h0 = es * s0, h1 = es * s1;
  float hn = fmaxf(sqrtf(wave_sum(h0 * h0 + h1 * h1)), kEps);
  if (hn > maxn) { float s = maxn / hn; h0 *= s; h1 *= s; }

  // xt = relu(logmap0(h))
  float ln = fmaxf(sqrtf(wave_sum(h0 * h0 + h1 * h1)), kEps);
  float ls = artanh_clip(ln) / ln;
  float t0 = fmaxf(ls * h0, 0.0f);
  float t1 = fmaxf(ls * h1, 0.0f);

  // out = proj(expmap0(xt)), c_out = 1
  float tn = fmaxf(sqrtf(wave_sum(t0 * t0 + t1 * t1)), kEps);
  float ts = tanhf(tn) / tn;
  float o0 = ts * t0, o1 = ts * t1;
  float on = fmaxf(sqrtf(wave_sum(o0 * o0 + o1 * o1)), kEps);
  if (on > maxn) { float s = maxn / on; o0 *= s; o1 *= s; }

  out[(size_t)row * DID + lane]      = o0;
  out[(size_t)row * DID + lane + 32] = o1;
}

// ---------------------------------------------------------------------------
extern "C" void kernel_launch(void* const* d_in, const int* in_sizes, int n_in,
                              void* d_out, int out_size, void* d_ws, size_t ws_size,
                              hipStream_t stream) {
  const float* x    = (const float*)d_in[0];
  const float* W    = (const float*)d_in[1];
  const float* b    = (const float*)d_in[2];
  const int*   erow = (const int*)d_in[3];
  const int*   ecol = (const int*)d_in[4];
  const float* evl  = (const float*)d_in[5];

  const int N = in_sizes[0] / DID;
  const int E = in_sizes[3];

  // ws layout: [0, N*64)   mx, later reused as the SpMM accumulator
  //            [N*64, 2N*64) xt (tangent features)
  float* mx_acc = (float*)d_ws;
  float* xt     = (float*)d_ws + (size_t)N * DID;
  const size_t nd = (size_t)N * DID;

  // 1) GEMM with fp32 WMMA
  gemm_wmma_f32<<<(N + 15) / 16, 128, 0, stream>>>(x, W, mx_acc, N);

  // 2) row-wise Mobius pipeline -> xt
  post_gemm_rowwise<<<(N + 7) / 8, 256, 0, stream>>>(x, mx_acc, b, xt, N);

  // 3) zero the accumulator (reuses mx region; runs after it is consumed)
  zero_f32<<<2048, 256, 0, stream>>>(mx_acc, nd);

  // 4) SpMM scatter with fp32 atomics (one wave / edge)
  spmm_atomic<<<(E + 7) / 8, 256, 0, stream>>>(erow, ecol, evl, xt, mx_acc, E);

  // 5) final activation chain -> d_out
  final_rowwise<<<(N + 7) / 8, 256, 0, stream>>>(mx_acc, (float*)d_out, N);
}